// SSAtt_bci_42202348650941
// MI455X (gfx1250) — compile-verified
//
#include <hip/hip_runtime.h>
#include <hip/hip_bf16.h>
#include <math.h>

typedef __attribute__((ext_vector_type(2))) float v2f;
typedef __attribute__((ext_vector_type(8))) float v8f;

#define NB     256
#define T_IN   1000
#define T_SIG  1001
#define NCH1   22
#define NCH2   25
#define NEP    3
#define NPATCH 30
#define NFEAT  513
#define K2CONV (NCH1 * 12)   // 264, reduction size of conv2

struct Ptrs { const float* sq[4]; const float* sk[4]; const float* sv[4]; };

__device__ inline v8f wmma4(v2f a, v2f b, v8f c) {
  return __builtin_amdgcn_wmma_f32_16x16x4_f32(false, a, false, b, (short)0, c, false, false);
}

// ---------------------------------------------------------------------------
// patch geometry: 30 patches over a 5x5 grid, k in {2,3,4,5}
// ---------------------------------------------------------------------------
__device__ inline void patch_decode(int i, int& gi, int& kk, int& k2, int& r0, int& c0) {
  int p;
  if (i < 16)      { gi = 0; p = i; }
  else if (i < 25) { gi = 1; p = i - 16; }
  else if (i < 29) { gi = 2; p = i - 25; }
  else             { gi = 3; p = 0; }
  kk = 2 + gi; k2 = kk * kk;
  int span = 6 - kk;
  r0 = p / span; c0 = p % span;
}

// ---------------------------------------------------------------------------
// per-thread 4x4 Jacobi eigensolver (fully register resident)
// ---------------------------------------------------------------------------
template<int P, int Q>
__device__ inline void rot4(float (&a)[4][4], float (&u)[4][4]) {
  float apq = a[P][Q];
  if (fabsf(apq) > 1e-20f) {
    float th = (a[Q][Q] - a[P][P]) / (2.f * apq);
    float t  = copysignf(1.f, th) / (fabsf(th) + sqrtf(1.f + th * th));
    float c  = 1.f / sqrtf(1.f + t * t);
    float s  = t * c;
#pragma unroll
    for (int k = 0; k < 4; ++k) { float akp = a[k][P], akq = a[k][Q]; a[k][P] = c*akp - s*akq; a[k][Q] = s*akp + c*akq; }
#pragma unroll
    for (int k = 0; k < 4; ++k) { float apk = a[P][k], aqk = a[Q][k]; a[P][k] = c*apk - s*aqk; a[P == Q ? P : Q][k] = s*apk + c*aqk; }
#pragma unroll
    for (int k = 0; k < 4; ++k) { float ukp = u[k][P], ukq = u[k][Q]; u[k][P] = c*ukp - s*ukq; u[k][Q] = s*ukp + c*ukq; }
  }
}

__device__ inline void jacobi4(float (&a)[4][4], float (&u)[4][4]) {
#pragma unroll
  for (int i = 0; i < 4; ++i)
#pragma unroll
    for (int j = 0; j < 4; ++j) u[i][j] = (i == j) ? 1.f : 0.f;
  for (int s = 0; s < 7; ++s) {
    rot4<0,1>(a,u); rot4<0,2>(a,u); rot4<0,3>(a,u);
    rot4<1,2>(a,u); rot4<1,3>(a,u); rot4<2,3>(a,u);
  }
}

__device__ inline void log4(const float (&a)[4][4], const float (&u)[4][4], float (&o)[4][4]) {
  float fl[4];
#pragma unroll
  for (int k = 0; k < 4; ++k) fl[k] = logf(fmaxf(a[k][k], 1e-9f));
#pragma unroll
  for (int i = 0; i < 4; ++i)
#pragma unroll
    for (int j = 0; j < 4; ++j) {
      float s = 0.f;
#pragma unroll
      for (int k = 0; k < 4; ++k) s += u[i][k] * fl[k] * u[j][k];
      o[i][j] = s;
    }
}

// ---------------------------------------------------------------------------
// wave-parallel cyclic Jacobi on an LDS-resident n x n symmetric matrix.
// Barrier-uniform: safe in multi-wave blocks when all waves use the same n and
// sweep count; `active=false` waves execute the barrier sequence only.
// ---------------------------------------------------------------------------
__device__ void wave_jacobi(float* A, float* U, int n, int sweeps, int lane, bool active) {
  if (active) for (int t = lane; t < n * n; t += 32) U[t] = (t / n == t % n) ? 1.f : 0.f;
  __syncthreads();
  for (int s = 0; s < sweeps; ++s) {
    for (int p = 0; p < n - 1; ++p) {
      for (int q = p + 1; q < n; ++q) {
        float c = 1.f, sn = 0.f;
        if (active) {
          float apq = A[p*n+q], app = A[p*n+p], aqq = A[q*n+q];
          if (fabsf(apq) > 1e-20f) {
            float th = (aqq - app) / (2.f * apq);
            float t  = copysignf(1.f, th) / (fabsf(th) + sqrtf(1.f + th * th));
            c  = 1.f / sqrtf(1.f + t * t);
            sn = t * c;
          }
        }
        __syncthreads();
        if (active && lane < n) {            // column rotation (lane owns row `lane`)
          float akp = A[lane*n+p], akq = A[lane*n+q];
          A[lane*n+p] = c*akp - sn*akq;
          A[lane*n+q] = sn*akp + c*akq;
        }
        __syncthreads();
        if (active && lane < n) {            // row rotation + eigenvector update
          float apk = A[p*n+lane], aqk = A[q*n+lane];
          A[p*n+lane] = c*apk - sn*aqk;
          A[q*n+lane] = sn*apk + c*aqk;
          float ukp = U[lane*n+p], ukq = U[lane*n+q];
          U[lane*n+p] = c*ukp - sn*ukq;
          U[lane*n+q] = sn*ukp + c*ukq;
        }
        __syncthreads();
      }
    }
  }
}

// out = U * f(diag(A)) * U^T ; mode 0: log(max(l,1e-9)), 1: max(exp(l),1e-4),
// 2: max(l, ln(1e-4))  (the fused rect∘exp and log∘rect∘exp transforms)
__device__ void wave_recon(const float* A, const float* U, float* out, float* flam,
                           int n, int lane, int mode, bool active) {
  if (active) for (int k = lane; k < n; k += 32) {
    float lm = A[k*n+k];
    float f;
    if (mode == 0)      f = logf(fmaxf(lm, 1e-9f));
    else if (mode == 1) f = fmaxf(expf(lm), 1e-4f);
    else                f = fmaxf(lm, -9.210340371976182f);
    flam[k] = f;
  }
  __syncthreads();
  if (active) for (int t = lane; t < n * n; t += 32) {
    int i = t / n, j = t % n;
    float s = 0.f;
    for (int k = 0; k < n; ++k) s += U[i*n+k] * flam[k] * U[j*n+k];
    out[t] = s;
  }
  __syncthreads();
}

// ---------------------------------------------------------------------------
// K1: conv1 (22x22 channel matmul) + BN1  -> y1[b][c][t]  (256x22x1000)
// ---------------------------------------------------------------------------
__global__ __launch_bounds__(256) void conv1_bn_kernel(
    const float* __restrict__ x, const float* __restrict__ w1, const float* __restrict__ b1,
    const float* __restrict__ g, const float* __restrict__ bb,
    const float* __restrict__ mm, const float* __restrict__ vv, float* __restrict__ y1) {
  int t = blockIdx.x * 256 + threadIdx.x;
  int c = blockIdx.y, b = blockIdx.z;
  if (t >= T_IN) return;
  const float* xb = x + (size_t)b * NCH1 * T_IN;
  float s = 0.f;
#pragma unroll
  for (int h = 0; h < NCH1; ++h) s += w1[c*NCH1 + h] * xb[h*T_IN + t];
  s += b1[c];
  float inv = g[c] * rsqrtf(vv[c] + 1e-5f);
  y1[((size_t)b * NCH1 + c) * T_IN + t] = (s - mm[c]) * inv + bb[c];
}

// ---------------------------------------------------------------------------
// K2: conv2 + BN2 as an implicit GEMM on the fp32 WMMA path.
//   C[o, t] = sum_{k=c*12+w} W[o, k] * y1[c, t+w-6]   (M=25->32, K=264, N=16/wave)
// Weights staged in LDS zero-padded to 32 rows (contiguous b64 A-fragments),
// y1 halo tile staged in LDS (B-fragments), BN folded to scale/shift.
// 8 waves/block, each owning a 16-wide t-tile; block covers 128 timesteps.
// ---------------------------------------------------------------------------
__global__ __launch_bounds__(256) void conv2_bn_wmma_kernel(
    const float* __restrict__ y1, const float* __restrict__ w2, const float* __restrict__ b2,
    const float* __restrict__ g, const float* __restrict__ bb,
    const float* __restrict__ mm, const float* __restrict__ vv, float* __restrict__ sig) {
  int b = blockIdx.y;
  int t0blk = blockIdx.x * 128;
  int tid = threadIdx.x;
  int wv = tid >> 5, lane = tid & 31;
  int hi = lane >> 4, ml = lane & 15;

  __shared__ float Wl[32 * K2CONV];   // Wl[m*264 + k], rows 25..31 zero
  __shared__ float Yt[NCH1 * 140];    // y1 tile, t-range [t0blk-6, t0blk+133)
  __shared__ float scale[NCH2], shift[NCH2];

  for (int e = tid; e < 32 * K2CONV; e += 256) {
    int m = e / K2CONV, k = e % K2CONV;
    Wl[e] = (m < NCH2) ? w2[m * K2CONV + k] : 0.f;
  }
  const float* yb = y1 + (size_t)b * NCH1 * T_IN;
  for (int e = tid; e < NCH1 * 140; e += 256) {
    int c = e / 140, tt = e % 140;
    int gidx = t0blk - 6 + tt;
    Yt[e] = (tt < 139 && gidx >= 0 && gidx < T_IN) ? yb[c * T_IN + gidx] : 0.f;
  }
  if (tid < NCH2) {
    float inv = g[tid] * rsqrtf(vv[tid] + 1e-5f);
    scale[tid] = inv;
    shift[tid] = (b2[tid] - mm[tid]) * inv + bb[tid];
  }
  __syncthreads();

  v8f acc0 = {}, acc1 = {};
  int tloc = 16 * wv + ml;                     // tile-local column base
  for (int k0 = 0; k0 < K2CONV; k0 += 4) {     // 66 branch-free WMMA k-steps
    int ka = k0 + 2 * hi;
    int kb = ka + 1;
    v2f a0, a1, bf;
    a0.x = Wl[ml * K2CONV + ka];
    a0.y = Wl[ml * K2CONV + kb];
    a1.x = Wl[(ml + 16) * K2CONV + ka];
    a1.y = Wl[(ml + 16) * K2CONV + kb];
    int ca = ka / 12, wa = ka % 12;
    int cb = kb / 12, wb = kb % 12;
    bf.x = Yt[ca * 140 + tloc + wa];
    bf.y = Yt[cb * 140 + tloc + wb];
    acc0 = wmma4(a0, bf, acc0);
    acc1 = wmma4(a1, bf, acc1);
  }
  int t = t0blk + 16 * wv + ml;
  if (t <= T_IN) {                             // T_SIG = 1001 outputs
    float* sg = sig + (size_t)b * NCH2 * T_SIG + t;
#pragma unroll
    for (int r = 0; r < 8; ++r) {
      int o = r + 8 * hi;
      sg[(size_t)o * T_SIG] = acc0[r] * scale[o] + shift[o];
      int o2 = o + 16;
      if (o2 < NCH2) sg[(size_t)o2 * T_SIG] = acc1[r] * scale[o2] + shift[o2];
    }
  }
}

// ---------------------------------------------------------------------------
// K3: segment covariance via fp32 WMMA Gram (one wave per (b,epoch))
//     cov = (X X^T - L mu mu^T)/(L-1), trace-normalized, +1e-5 I
// Main K-loop is branch-free (clamped row + zero mask); ragged K tail guarded.
// ---------------------------------------------------------------------------
__global__ __launch_bounds__(32) void cov_kernel(const float* __restrict__ sig,
                                                 float* __restrict__ covs) {
  int be = blockIdx.x;
  int b = be / NEP, ep = be % NEP;
  const int offs[3] = {0, 334, 668};
  const int lens[3] = {334, 334, 333};
  int off = offs[ep], L = lens[ep];
  int lane = threadIdx.x;
  const float* S = sig + (size_t)b * NCH2 * T_SIG + off;
  __shared__ float mu[32];
  __shared__ float G[32][32];
  __shared__ float covl[625];
  __shared__ float trs;
  float m = 0.f;
  if (lane < 25) {
    for (int t = 0; t < L; ++t) m += S[lane*T_SIG + t];
    m /= (float)L;
  }
  mu[lane] = (lane < 25) ? m : 0.f;
  // 32x32 Gram as 2x2 grid of 16x16 WMMA tiles; Gram => A-frag == B-frag.
  v8f c00 = {}, c01 = {}, c10 = {}, c11 = {};
  int hi = lane >> 4, ml = lane & 15;
  int r1 = ml + 16;
  int r1c = (r1 < 25) ? r1 : 0;
  float mask1 = (r1 < 25) ? 1.f : 0.f;
  int Lmain = L & ~3;
  for (int k0 = 0; k0 < Lmain; k0 += 4) {
    int ka = k0 + 2*hi;
    v2f f0, f1;
    f0.x = S[ml*T_SIG + ka];
    f0.y = S[ml*T_SIG + ka + 1];
    f1.x = mask1 * S[r1c*T_SIG + ka];
    f1.y = mask1 * S[r1c*T_SIG + ka + 1];
    c00 = wmma4(f0, f0, c00);
    c01 = wmma4(f0, f1, c01);
    c10 = wmma4(f1, f0, c10);
    c11 = wmma4(f1, f1, c11);
  }
  if (Lmain < L) {                     // ragged tail (L%4 = 1 or 2)
    int ka = Lmain + 2*hi, kb = ka + 1;
    v2f f0, f1;
    f0.x = (ka < L) ? S[ml*T_SIG + ka] : 0.f;
    f0.y = (kb < L) ? S[ml*T_SIG + kb] : 0.f;
    f1.x = (ka < L) ? mask1 * S[r1c*T_SIG + ka] : 0.f;
    f1.y = (kb < L) ? mask1 * S[r1c*T_SIG + kb] : 0.f;
    c00 = wmma4(f0, f0, c00);
    c01 = wmma4(f0, f1, c01);
    c10 = wmma4(f1, f0, c10);
    c11 = wmma4(f1, f1, c11);
  }
#pragma unroll
  for (int r = 0; r < 8; ++r) {
    G[r + 8*hi][ml]           = c00[r];
    G[r + 8*hi][16 + ml]      = c01[r];
    G[16 + r + 8*hi][ml]      = c10[r];
    G[16 + r + 8*hi][16 + ml] = c11[r];
  }
  __syncthreads();
  float inv = 1.f / (float)(L - 1);
  for (int t = lane; t < 625; t += 32) {
    int i = t / 25, j = t % 25;
    covl[t] = (G[i][j] - (float)L * mu[i] * mu[j]) * inv;
  }
  __syncthreads();
  if (lane == 0) {
    float tr = 0.f;
    for (int d = 0; d < 25; ++d) tr += covl[d*25 + d];
    trs = tr;
  }
  __syncthreads();
  float itr = 1.f / trs;
  for (int t = lane; t < 625; t += 32) {
    int i = t / 25, j = t % 25;
    covs[(size_t)be*625 + t] = covl[t] * itr + ((i == j) ? 1e-5f : 0.f);
  }
}

// ---------------------------------------------------------------------------
// K4: attention weights. Only column -1 of the softmax is consumed by the
// reference, so we need logK for all 30 patches but logQ only for patch 29.
// One wave per (b,epoch); one 4x4 eigensolve per lane.
// ---------------------------------------------------------------------------
__device__ void bimap_small(const float* xb, const float* __restrict__ W,
                            int kk, int k2, int r0, int c0, float* tcol, float (&out)[4][4]) {
#pragma unroll
  for (int c = 0; c < 4; ++c) {
    for (int j1 = 0; j1 < k2; ++j1) {
      int g1 = (r0 + j1/kk)*5 + (c0 + j1%kk);
      float s = 0.f;
      for (int j2 = 0; j2 < k2; ++j2) {
        int g2 = (r0 + j2/kk)*5 + (c0 + j2%kk);
        s += xb[g1*25 + g2] * W[j2*4 + c];
      }
      tcol[j1] = s;
    }
#pragma unroll
    for (int r = 0; r < 4; ++r) {
      float s = 0.f;
      for (int j1 = 0; j1 < k2; ++j1) s += W[j1*4 + r] * tcol[j1];
      out[r][c] = s;
    }
  }
}

__global__ __launch_bounds__(32) void weights_kernel(const float* __restrict__ covs, Ptrs pp,
                                                     float* __restrict__ wts) {
  int be = blockIdx.x;
  int lane = threadIdx.x;
  __shared__ float xb[625];
  __shared__ float tls[32 * 25];
  __shared__ float logqS[16];
  __shared__ float qnS;
  __shared__ float scS[32];
  __shared__ float mxS, smS;
  for (int t = lane; t < 625; t += 32) xb[t] = covs[(size_t)be*625 + t];
  __syncthreads();
  float lgK[4][4] = {};
  float kn = 0.f;
  int i = lane;
  if (i < NPATCH) {
    int gi, kk, k2, r0, c0;
    patch_decode(i, gi, kk, k2, r0, c0);
    float M[4][4], Uu[4][4];
    bimap_small(xb, pp.sk[gi], kk, k2, r0, c0, &tls[lane*25], M);
#pragma unroll
    for (int d = 0; d < 4; ++d) M[d][d] += 2.5e-5f * (d + 1);   // diag_jitter, n=4
    jacobi4(M, Uu);
    log4(M, Uu, lgK);
#pragma unroll
    for (int a = 0; a < 4; ++a)
#pragma unroll
      for (int bcl = 0; bcl < 4; ++bcl) kn += lgK[a][bcl] * lgK[a][bcl];
    if (i == 29) {            // logQ only needed for the last (k=5) patch
      float Mq[4][4], Uq[4][4], lgQ[4][4];
      bimap_small(xb, pp.sq[3], 5, 25, 0, 0, &tls[lane*25], Mq);
#pragma unroll
      for (int d = 0; d < 4; ++d) Mq[d][d] += 2.5e-5f * (d + 1);
      jacobi4(Mq, Uq);
      log4(Mq, Uq, lgQ);
      float qn = 0.f;
#pragma unroll
      for (int a = 0; a < 4; ++a)
#pragma unroll
        for (int bcl = 0; bcl < 4; ++bcl) { logqS[a*4 + bcl] = lgQ[a][bcl]; qn += lgQ[a][bcl]*lgQ[a][bcl]; }
      qnS = qn;
    }
  }
  __syncthreads();
  float sc = -1e30f;
  if (i < NPATCH) {
    float cross = 0.f;
#pragma unroll
    for (int a = 0; a < 4; ++a)
#pragma unroll
      for (int bcl = 0; bcl < 4; ++bcl) cross += lgK[a][bcl] * logqS[a*4 + bcl];
    float E = fmaxf(kn + qnS - 2.f * cross, 0.f);
    sc = 1.f / (1.f + log1pf(E));
  }
  scS[lane] = sc;
  __syncthreads();
  if (lane == 0) {
    float mx = -1e30f;
    for (int k = 0; k < NPATCH; ++k) mx = fmaxf(mx, scS[k]);
    float sm = 0.f;
    for (int k = 0; k < NPATCH; ++k) sm += expf(scS[k] - mx);
    mxS = mx; smS = sm;
  }
  __syncthreads();
  if (i < NPATCH) wts[(size_t)be*NPATCH + i] = expf(sc - mxS) / smS;
}

// ---------------------------------------------------------------------------
// K5: fused per-(b,epoch) branch: 30x [bimap V (25x25) -> jitter -> Jacobi ->
// logV -> acc += w_i * logV_i] with 8 waves x 4 patch-slots, then the 25x25
// eigensolve of out_log with rect∘exp fused into the eigenvalue transform.
// Deterministic (register accumulators, fixed reduction order, no atomics).
// ---------------------------------------------------------------------------
__device__ void bimap25(const float* xb, const float* __restrict__ sv, float* T, float* A,
                        int kk, int k2, int r0, int c0, int lane, bool active) {
  if (active) for (int t = lane; t < k2 * 25; t += 32) {
    int j1 = t / 25, c = t % 25;
    int g1 = (r0 + j1/kk)*5 + (c0 + j1%kk);
    float s = 0.f;
    for (int j2 = 0; j2 < k2; ++j2) {
      int g2 = (r0 + j2/kk)*5 + (c0 + j2%kk);
      s += xb[g1*25 + g2] * sv[j2*25 + c];
    }
    T[t] = s;
  }
  __syncthreads();
  if (active) for (int t = lane; t < 625; t += 32) {
    int r = t / 25, c = t % 25;
    float s = 0.f;
    for (int j1 = 0; j1 < k2; ++j1) s += sv[j1*25 + r] * T[j1*25 + c];
    A[t] = s;
  }
  __syncthreads();
}

__global__ __launch_bounds__(256) void branch_kernel(const float* __restrict__ covs,
                                                     const float* __restrict__ wts,
                                                     Ptrs pp, float* __restrict__ xm) {
  int be = blockIdx.x;
  int tid = threadIdx.x;
  int w = tid >> 5, lane = tid & 31;
  __shared__ float xb[625];
  __shared__ float Aw[8][625];
  __shared__ float Uw[8][625];      // doubles as bimap scratch (never live together)
  __shared__ float flamw[8][25];
  __shared__ float flamF[25];
  for (int t = tid; t < 625; t += 256) xb[t] = covs[(size_t)be*625 + t];
  __syncthreads();
  float acc[20];
#pragma unroll
  for (int idx = 0; idx < 20; ++idx) acc[idx] = 0.f;
  for (int iter = 0; iter < 4; ++iter) {
    int i = iter * 8 + w;
    bool active = (i < NPATCH);
    int gi = 0, kk = 2, k2 = 4, r0 = 0, c0 = 0;
    if (active) patch_decode(i, gi, kk, k2, r0, c0);
    bimap25(xb, pp.sv[gi], Uw[w], Aw[w], kk, k2, r0, c0, lane, active);
    if (active && lane < 25) Aw[w][lane*25 + lane] += 4e-6f * (lane + 1);  // jitter, n=25
    __syncthreads();
    wave_jacobi(Aw[w], Uw[w], 25, 10, lane, active);
    if (active) for (int k = lane; k < 25; k += 32)
      flamw[w][k] = logf(fmaxf(Aw[w][k*25 + k], 1e-9f));
    __syncthreads();
    float wv = active ? wts[(size_t)be*NPATCH + i] : 0.f;
    if (active) {
#pragma unroll
      for (int idx = 0; idx < 20; ++idx) {
        int t = lane + idx * 32;
        if (t < 625) {
          int ii = t / 25, jj = t % 25;
          float s = 0.f;
          for (int k = 0; k < 25; ++k) s += Uw[w][ii*25 + k] * flamw[w][k] * Uw[w][jj*25 + k];
          acc[idx] += wv * s;
        }
      }
    }
    __syncthreads();
  }
  // stage register accumulators and reduce across waves (fixed order)
#pragma unroll
  for (int idx = 0; idx < 20; ++idx) {
    int t = lane + idx * 32;
    if (t < 625) Aw[w][t] = acc[idx];
  }
  __syncthreads();
  for (int t = tid; t < 625; t += 256) {
    int ii = t / 25, jj = t % 25;
    float s1 = 0.f, s2 = 0.f;
    for (int ww = 0; ww < 8; ++ww) { s1 += Aw[ww][t]; s2 += Aw[ww][jj*25 + ii]; }
    xb[t] = 0.5f * (s1 + s2);          // symmetrized out_log
  }
  __syncthreads();
  wave_jacobi(xb, Uw[0], 25, 10, lane, w == 0);
  if (tid < 25) flamF[tid] = fmaxf(expf(xb[tid*25 + tid]), 1e-4f);  // rect∘exp fused
  __syncthreads();
  for (int t = tid; t < 625; t += 256) {
    int ii = t / 25, jj = t % 25;
    float s = 0.f;
    for (int k = 0; k < 25; ++k) s += Uw[0][ii*25 + k] * flamF[k] * Uw[0][jj*25 + k];
    xm[(size_t)be*625 + t] = s;
  }
}

// ---------------------------------------------------------------------------
// K6: second SPD attention over the 3 epochs (18x18), one block per batch,
// 3 waves (one epoch each), then triu-feature extraction.
// ---------------------------------------------------------------------------
__device__ void bimap18(const float* X, const float* __restrict__ W, float* tmp, float* out, int lane) {
  for (int t = lane; t < 450; t += 32) {
    int j = t / 18, c = t % 18;
    float s = 0.f;
    for (int j2 = 0; j2 < 25; ++j2) s += X[j*25 + j2] * W[j2*18 + c];
    tmp[t] = s;
  }
  __syncthreads();
  for (int t = lane; t < 324; t += 32) {
    int r = t / 18, c = t % 18;
    float s = 0.f;
    for (int j = 0; j < 25; ++j) s += W[j*18 + r] * tmp[j*18 + c];
    out[t] = s;
  }
  __syncthreads();
}

__global__ __launch_bounds__(96) void attn2_kernel(const float* __restrict__ xm,
    const float* __restrict__ mq, const float* __restrict__ mk, const float* __restrict__ mv,
    float* __restrict__ feats) {
  int b = blockIdx.x;
  int tid = threadIdx.x;
  int w = tid >> 5, lane = tid & 31;
  __shared__ float xs[NEP * 625];
  __shared__ float tmpw[NEP][450];
  __shared__ float Aw[NEP][324];
  __shared__ float Uw[NEP][324];
  __shared__ float flamw[NEP][20];
  __shared__ float lq[NEP][324], lk[NEP][324], lv[NEP][324];
  __shared__ float qn[NEP], kn[NEP];
  __shared__ float red[NEP][32];
  __shared__ float scm[9];
  __shared__ float Pm[NEP][NEP];
  for (int t = tid; t < NEP * 625; t += 96) xs[t] = xm[(size_t)b * NEP * 625 + t];
  __syncthreads();
  const float* X = &xs[w * 625];
  // ---- logQ ----
  bimap18(X, mq, tmpw[w], Aw[w], lane);
  wave_jacobi(Aw[w], Uw[w], 18, 9, lane, true);
  wave_recon(Aw[w], Uw[w], lq[w], flamw[w], 18, lane, 0, true);
  { float s = 0.f; for (int t = lane; t < 324; t += 32) { float v0 = lq[w][t]; s += v0*v0; } red[w][lane] = s; }
  __syncthreads();
  if (lane == 0) { float s = 0.f; for (int k = 0; k < 32; ++k) s += red[w][k]; qn[w] = s; }
  __syncthreads();
  // ---- logK ----
  bimap18(X, mk, tmpw[w], Aw[w], lane);
  wave_jacobi(Aw[w], Uw[w], 18, 9, lane, true);
  wave_recon(Aw[w], Uw[w], lk[w], flamw[w], 18, lane, 0, true);
  { float s = 0.f; for (int t = lane; t < 324; t += 32) { float v0 = lk[w][t]; s += v0*v0; } red[w][lane] = s; }
  __syncthreads();
  if (lane == 0) { float s = 0.f; for (int k = 0; k < 32; ++k) s += red[w][k]; kn[w] = s; }
  __syncthreads();
  // ---- logV ----
  bimap18(X, mv, tmpw[w], Aw[w], lane);
  wave_jacobi(Aw[w], Uw[w], 18, 9, lane, true);
  wave_recon(Aw[w], Uw[w], lv[w], flamw[w], 18, lane, 0, true);
  __syncthreads();
  // ---- scores / softmax (axis=-2) ----
  if (tid < 9) {
    int i = tid / 3, j = tid % 3;
    float cr = 0.f;
    for (int t = 0; t < 324; ++t) cr += lk[i][t] * lq[j][t];
    float E = fmaxf(kn[i] + qn[j] - 2.f * cr, 0.f);
    scm[tid] = 1.f / (1.f + log1pf(E));
  }
  __syncthreads();
  if (tid < 3) {
    int j = tid;
    float mx = fmaxf(scm[j], fmaxf(scm[3 + j], scm[6 + j]));
    float e0 = expf(scm[j] - mx), e1 = expf(scm[3 + j] - mx), e2 = expf(scm[6 + j] - mx);
    float den = e0 + e1 + e2;
    Pm[j][0] = e0 / den; Pm[j][1] = e1 / den; Pm[j][2] = e2 / den;
  }
  __syncthreads();
  // ---- out_log_w = sum_i P[w][i] * logV_i, symmetrize ----
  for (int t = lane; t < 324; t += 32)
    tmpw[w][t] = Pm[w][0] * lv[0][t] + Pm[w][1] * lv[1][t] + Pm[w][2] * lv[2][t];
  __syncthreads();
  for (int t = lane; t < 324; t += 32) {
    int i = t / 18, j = t % 18;
    Aw[w][t] = 0.5f * (tmpw[w][t] + tmpw[w][j*18 + i]);
  }
  __syncthreads();
  // ---- Lg = U max(lambda, ln 1e-4) U^T  (log∘rect∘exp fused) ----
  wave_jacobi(Aw[w], Uw[w], 18, 9, lane, true);
  wave_recon(Aw[w], Uw[w], lq[w], flamw[w], 18, lane, 2, true);
  for (int t = lane; t < 324; t += 32) {
    int i = t / 18, j = t % 18;
    if (j >= i) {
      int idx = 18*i - (i*(i - 1)) / 2 + (j - i);
      float coef = (i == j) ? 1.f : 1.4142135623730951f;
      feats[(size_t)b * NFEAT + w * 171 + idx] = coef * lq[w][t];
    }
  }
}

// ---------------------------------------------------------------------------
// K7: final linear 513 -> 4
// ---------------------------------------------------------------------------
__global__ __launch_bounds__(256) void linear_kernel(const float* __restrict__ feats,
    const float* __restrict__ lw, const float* __restrict__ lb, float* __restrict__ out) {
  int t = blockIdx.x * 256 + threadIdx.x;
  if (t >= NB * 4) return;
  int b = t >> 2, r = t & 3;
  float s = lb[r];
  for (int f = 0; f < NFEAT; ++f) s += feats[(size_t)b*NFEAT + f] * lw[r*NFEAT + f];
  out[t] = s;
}

// ---------------------------------------------------------------------------
extern "C" void kernel_launch(void* const* d_in, const int* in_sizes, int n_in,
                              void* d_out, int out_size, void* d_ws, size_t ws_size,
                              hipStream_t stream) {
  (void)in_sizes; (void)n_in; (void)out_size; (void)ws_size;
  const float* x    = (const float*)d_in[0];
  const float* c1w  = (const float*)d_in[1];
  const float* c1b  = (const float*)d_in[2];
  const float* bn1g = (const float*)d_in[3];
  const float* bn1b = (const float*)d_in[4];
  const float* bn1m = (const float*)d_in[5];
  const float* bn1v = (const float*)d_in[6];
  const float* c2w  = (const float*)d_in[7];
  const float* c2b  = (const float*)d_in[8];
  const float* bn2g = (const float*)d_in[9];
  const float* bn2b = (const float*)d_in[10];
  const float* bn2m = (const float*)d_in[11];
  const float* bn2v = (const float*)d_in[12];
  Ptrs pp;
  for (int i = 0; i < 4; ++i) {
    pp.sq[i] = (const float*)d_in[13 + i];
    pp.sk[i] = (const float*)d_in[17 + i];
    pp.sv[i] = (const float*)d_in[21 + i];
  }
  const float* mq = (const float*)d_in[25];
  const float* mk = (const float*)d_in[26];
  const float* mv = (const float*)d_in[27];
  const float* lw = (const float*)d_in[28];
  const float* lb = (const float*)d_in[29];

  float* ws = (float*)d_ws;
  // peak-live layout: y1 + sig concurrently; small buffers overlay y1 afterwards
  float* y1    = ws;                    // 256*22*1000 = 5,632,000 f
  float* sig   = ws + 5632000;          // 256*25*1001 = 6,406,400 f
  float* covs  = ws;                    // 480,000 f (y1 dead after conv2)
  float* wtsb  = ws + 480000;           //  23,040 f
  float* xmb   = ws + 503040;           // 480,000 f
  float* feats = ws + 983040;           // 131,328 f

  conv1_bn_kernel<<<dim3(4, NCH1, NB), 256, 0, stream>>>(x, c1w, c1b, bn1g, bn1b, bn1m, bn1v, y1);
  conv2_bn_wmma_kernel<<<dim3(8, NB), 256, 0, stream>>>(y1, c2w, c2b, bn2g, bn2b, bn2m, bn2v, sig);
  cov_kernel<<<NB * NEP, 32, 0, stream>>>(sig, covs);
  weights_kernel<<<NB * NEP, 32, 0, stream>>>(covs, pp, wtsb);
  branch_kernel<<<NB * NEP, 256, 0, stream>>>(covs, wtsb, pp, xmb);
  attn2_kernel<<<NB, 96, 0, stream>>>(xmb, mq, mk, mv, feats);
  linear_kernel<<<4, 256, 0, stream>>>(feats, lw, lb, (float*)d_out);
}